// ModelNew_58798102282513
// MI455X (gfx1250) — compile-verified
//
#include <hip/hip_runtime.h>
#include <hip/hip_bf16.h>

// ---- problem constants (from reference) ----
#define NB    16
#define CIN   32
#define CO    64
#define DD    16
#define HH    64
#define WW    64
#define DOUT  14
#define HOUT  62
#define WOUT  62

#define NT    512            // threads per block = 16 waves (wave32)
#define KQ    288            // K-quads: ci(32) * kd(3) * kh(3); kw padded 3->4
#define WPAD  68             // x slab row width (needs 67, pad to 68)
#define WROW  65             // weight LDS row pitch (bank-conflict-free; col 64 unused)
#define WK    864            // real K = 32*27
#define WLDS  (WK * WROW)    // 56160 floats of weights in LDS
#define XSLAB (KQ * WPAD)    // 19584 floats of x slab
#define LDS_FLOATS (WLDS + XSLAB)   // 75744 floats = 302976 B  (< 320KB WGP LDS)

typedef __attribute__((ext_vector_type(2))) float v2f;
typedef __attribute__((ext_vector_type(8))) float v8f;

__global__ void conv3d_min_softmax_kernel(const float* __restrict__ x,
                                          const float* __restrict__ wgt,
                                          const float* __restrict__ bias,
                                          float* __restrict__ out) {
  extern __shared__ float lds[];          // [0,WLDS): weights  [WLDS,..): x slab / ymin
  float* Xl = lds + WLDS;

  const int tid    = threadIdx.x;
  const int wid    = tid >> 5;
  const int lane   = tid & 31;
  const int lane_m = lane & 15;
  const int khalf  = lane >> 4;           // 0: K-offsets {0,1}, 1: {2,3}
  const int kb0    = khalf * 2;

  const int mt  = wid & 3;                // M (wo) sub-tile 0..3
  const int ct  = wid >> 2;               // N (co) sub-tile 0..3
  const int wo0 = mt * 16;
  const int co0 = ct * 16;

  const int blk = blockIdx.x;             // 0..N*HOUT-1
  const int n   = blk / HOUT;
  const int ho  = blk - n * HOUT;

  // ---- stage all weights into LDS once: Wlds[k*WROW + co] = wgt[co*WK + k] ----
  // coalesced global reads, conflict-free LDS stores (pitch 65, co -> distinct banks)
  for (int i = tid; i < CO * WK; i += NT) {
    int co = i / WK;
    int k  = i - co * WK;
    lds[k * WROW + co] = wgt[i];
  }
  if (tid == 0) lds[64] = 0.0f;           // dedicated zero slot (row 0, unused col 64)

  float vmin[8];
#pragma unroll
  for (int r = 0; r < 8; ++r) vmin[r] = 3.402823466e38f;

  // Per-lane running LDS addresses (dword indices), branchless zero-padding:
  //   A lane element (M=lane_m, K-off kb0 / kb0+1): Xl[q*WPAD + wo0+lane_m+kb0 + {0,1}]
  //   B lane element (N=lane_m, K-off kb0):         lds[(q*3+kb0)*WROW + co0+lane_m]
  //   B lane element (K-off kb0+1): real row for lanes 0-15 (kw=1); the zero slot
  //   for lanes 16-31 (kw=3 is the pad) -> per-lane base+stride, no select needed.
  const int abase0 = wo0 + lane_m + kb0;                       // step WPAD
  const int bx0    = kb0 * WROW + co0 + lane_m;                // step 3*WROW
  const int by0    = (khalf == 0) ? (bx0 + WROW) : 64;         // step byst
  const int byst   = (khalf == 0) ? (3 * WROW) : 0;

  for (int d0 = 0; d0 < DOUT; ++d0) {
    __syncthreads();  // previous compute done before restaging slab

    // ---- stage x slab: row r=(ci*9+kd*3+kh) holds x[n][ci][d0+kd][ho+kh][0..63], pad w>=64 with 0
    for (int r = tid >> 4; r < KQ; r += 32) {
      int ci  = r / 9;
      int rem = r - ci * 9;
      int kd  = rem / 3;
      int kh  = rem - kd * 3;
      const float* src = x + (((n * CIN + ci) * DD + (d0 + kd)) * HH + (ho + kh)) * WW;
      __builtin_prefetch(src + HH * WW, 0, 0);    // next depth slice -> global_prefetch
      float* dst = Xl + r * WPAD;
      for (int w = (tid & 15); w < WPAD; w += 16)
        dst[w] = (w < WW) ? src[w] : 0.0f;
    }
    __syncthreads();

    // ---- implicit GEMM over K' = 288 quads via V_WMMA_F32_16X16X4_F32 ----
    v8f acc = {};
    int abase = abase0;
    int bx    = bx0;
    int by    = by0;
    for (int q = 0; q < KQ; ++q) {
      v2f a;
      a.x = Xl[abase];                // ds_load_2addr_b32 (offset1:1)
      a.y = Xl[abase + 1];
      v2f b;
      b.x = lds[bx];                  // always a real weight row
      b.y = lds[by];                  // kw=1 row (lanes 0-15) or zero slot (lanes 16-31)
      acc = __builtin_amdgcn_wmma_f32_16x16x4_f32(
          false, a, false, b, (short)0, acc, false, false);
      abase += WPAD;
      bx    += 3 * WROW;
      by    += byst;
    }
#pragma unroll
    for (int r = 0; r < 8; ++r) vmin[r] = fminf(vmin[r], acc[r]);
  }

  // ---- dump per-wave 16x16 min tiles to LDS (alias x slab): ymin[wo][co] ----
  __syncthreads();
  float* ymin = Xl;                       // 64*64 floats
  float* smx  = Xl + 4096;                // 64
  float* sinv = Xl + 4160;                // 64
#pragma unroll
  for (int r = 0; r < 8; ++r) {
    int m = wo0 + r + khalf * 8;          // C/D layout: lanes 0-15 -> M=r, 16-31 -> M=r+8
    ymin[m * 64 + (co0 + lane_m)] = vmin[r];
  }
  __syncthreads();

  // ---- channel softmax of (min + bias), one thread per pixel for max/sum ----
  if (tid < 64) {
    int wo = tid;
    float mx = -3.402823466e38f;
    for (int co = 0; co < CO; ++co)
      mx = fmaxf(mx, ymin[wo * 64 + co] + bias[co]);
    float s = 0.0f;
    for (int co = 0; co < CO; ++co)
      s += __expf(ymin[wo * 64 + co] + bias[co] - mx);
    smx[wo]  = mx;
    sinv[wo] = 1.0f / s;
  }
  __syncthreads();

  // ---- write out[n][co][ho][wo], coalesced along wo ----
  for (int i = tid; i < CO * 64; i += NT) {
    int wo = i & 63;
    int co = i >> 6;
    if (wo < WOUT) {
      float v = __expf(ymin[wo * 64 + co] + bias[co] - smx[wo]) * sinv[wo];
      out[((n * CO + co) * HOUT + ho) * WOUT + wo] = v;
    }
  }
}

extern "C" void kernel_launch(void* const* d_in, const int* in_sizes, int n_in,
                              void* d_out, int out_size, void* d_ws, size_t ws_size,
                              hipStream_t stream) {
  const float* x    = (const float*)d_in[0];
  const float* wgt  = (const float*)d_in[1];
  const float* bias = (const float*)d_in[2];
  float* out        = (float*)d_out;
  (void)in_sizes; (void)n_in; (void)out_size; (void)d_ws; (void)ws_size;

  dim3 grid(NB * HOUT, 1, 1);   // one block per (n, ho) output row
  dim3 block(NT, 1, 1);
  size_t smem = LDS_FLOATS * sizeof(float);  // 302976 B
  conv3d_min_softmax_kernel<<<grid, block, smem, stream>>>(x, wgt, bias, out);
}